// SingleObsEncoder_67791763800610
// MI455X (gfx1250) — compile-verified
//
#include <hip/hip_runtime.h>

#define N_NODES 20000
#define N_EDGES 320000
#define F_INPUT 7
#define HEADS   4
#define C1      128
#define HC1     512   // HEADS*C1
#define C2      64
#define NEG_SLOPE 0.2f

typedef __attribute__((ext_vector_type(2))) float v2f;
typedef __attribute__((ext_vector_type(8))) float v8f;

__device__ __forceinline__ float lrelu(float v) { return v > 0.f ? v : NEG_SLOPE * v; }
__device__ __forceinline__ float elu_f(float v) { return v > 0.f ? v : expm1f(v); }
// monotonic float->int ordering for atomicMax on signed ints
__device__ __forceinline__ int   f2ord(float f) { int i = __float_as_int(f); return i >= 0 ? i : (i ^ 0x7FFFFFFF); }
__device__ __forceinline__ float ord2f(int i)   { return __int_as_float(i >= 0 ? i : (i ^ 0x7FFFFFFF)); }

__global__ void fill_f32_kernel(float* p, float v, int n) {
    int i = blockIdx.x * blockDim.x + threadIdx.x;
    if (i < n) p[i] = v;
}
__global__ void fill_i32_kernel(int* p, int v, int n) {
    int i = blockIdx.x * blockDim.x + threadIdx.x;
    if (i < n) p[i] = v;
}

// -------- layer 1 linear: xl1 = x@Wl1+bl1, xr1 = x@Wr1+br1 (K=7, scalar) --------
__global__ void gemm1_kernel(const float* __restrict__ x,
                             const float* __restrict__ Wl, const float* __restrict__ bl,
                             const float* __restrict__ Wr, const float* __restrict__ br,
                             float* __restrict__ xl, float* __restrict__ xr) {
    int i = blockIdx.x * blockDim.x + threadIdx.x;   // over N_NODES*HC1
    if (i >= N_NODES * HC1) return;
    int n = i / HC1, c = i % HC1;
    const float* xrow = x + n * F_INPUT;
    float al = bl[c], ar = br[c];
#pragma unroll
    for (int f = 0; f < F_INPUT; ++f) {
        float xv = xrow[f];
        al += xv * Wl[f * HC1 + c];
        ar += xv * Wr[f * HC1 + c];
    }
    xl[i] = al; xr[i] = ar;
}

// -------- layer 1 edge logits: one wave per edge, 4 heads --------
__global__ void edge_logits1_kernel(const float* __restrict__ xl, const float* __restrict__ xr,
                                    const int* __restrict__ src, const int* __restrict__ dst,
                                    const float* __restrict__ att, float* __restrict__ e1,
                                    int* __restrict__ emax_ord) {
    int warp = (blockIdx.x * blockDim.x + threadIdx.x) >> 5;
    int lane = threadIdx.x & 31;
    if (warp >= N_EDGES) return;
    int s = src[warp], d = dst[warp];
    const float* pl = xl + (long)s * HC1;
    const float* pr = xr + (long)d * HC1;
    float acc[HEADS] = {0.f, 0.f, 0.f, 0.f};
    for (int i = lane; i < HC1; i += 32) {
        int h = i >> 7;                       // C1 = 128
        acc[h] += lrelu(pl[i] + pr[i]) * att[i];
    }
#pragma unroll
    for (int h = 0; h < HEADS; ++h) {
        float a = acc[h];
        for (int off = 16; off > 0; off >>= 1) a += __shfl_xor(a, off, 32);
        if (lane == 0) {
            e1[(long)warp * HEADS + h] = a;
            atomicMax(emax_ord + d * HEADS + h, f2ord(a));
        }
    }
}

__global__ void finalize_max_kernel(const int* __restrict__ ord, float* __restrict__ emaxf, int n) {
    int i = blockIdx.x * blockDim.x + threadIdx.x;
    if (i >= n) return;
    int o = ord[i];
    emaxf[i] = (o == (int)0x80000000) ? 0.f : ord2f(o);   // no incoming edges -> 0 (ref semantics)
}

// exp(e - emax[dst]) stored back into e, denom accumulated
__global__ void edge_exp_kernel(const int* __restrict__ dst, const float* __restrict__ emaxf,
                                float* __restrict__ e, float* __restrict__ denom, int nE, int H) {
    int i = blockIdx.x * blockDim.x + threadIdx.x;
    if (i >= nE * H) return;
    int eidx = i / H, h = i - eidx * H;
    int d = dst[eidx];
    float ee = expf(e[i] - emaxf[d * H + h]);
    e[i] = ee;
    atomicAdd(denom + d * H + h, ee);
}

// -------- layer 1 scatter: one wave per edge, atomic add alpha*xl[src] into out[dst] --------
__global__ void edge_scatter1_kernel(const float* __restrict__ xl,
                                     const int* __restrict__ src, const int* __restrict__ dst,
                                     const float* __restrict__ ee, const float* __restrict__ denom,
                                     float* __restrict__ out) {
    int warp = (blockIdx.x * blockDim.x + threadIdx.x) >> 5;
    int lane = threadIdx.x & 31;
    if (warp >= N_EDGES) return;
    int s = src[warp], d = dst[warp];
    const float* pl = xl + (long)s * HC1;
    float* po = out + (long)d * HC1;
#pragma unroll
    for (int h = 0; h < HEADS; ++h) {
        float alpha = ee[(long)warp * HEADS + h] / (denom[d * HEADS + h] + 1e-16f);
        for (int c = lane; c < C1; c += 32)
            atomicAdd(po + h * C1 + c, alpha * pl[h * C1 + c]);
    }
}

__global__ void elu_bias1_kernel(float* __restrict__ out1, const float* __restrict__ b1) {
    int i = blockIdx.x * blockDim.x + threadIdx.x;
    if (i >= N_NODES * HC1) return;
    out1[i] = elu_f(out1[i] + b1[i % HC1]);
}

// -------- layer 2 linear via fp32 WMMA: xl2 = h@Wl2+bl2, xr2 = h@Wr2+br2 --------
// one wave per 16x16 output tile; M=20000=1250*16, N=64=4*16, K=512 (128 k-steps of 4)
__global__ void gemm2_wmma_kernel(const float* __restrict__ A,
                                  const float* __restrict__ Wl, const float* __restrict__ bl,
                                  const float* __restrict__ Wr, const float* __restrict__ br,
                                  float* __restrict__ xl2, float* __restrict__ xr2) {
    int wave = (blockIdx.x * blockDim.x + threadIdx.x) >> 5;
    int lane = threadIdx.x & 31;
    int nt = wave & 3;          // 4 n-tiles
    int mt = wave >> 2;         // 1250 m-tiles
    int m0 = mt * 16, n0 = nt * 16;
    int half = lane >> 4, l16 = lane & 15;
    // A fragment (16x4 f32): lanes 0-15 hold row=l16 K={k0,k0+1}; lanes 16-31 hold K={k0+2,k0+3}
    const float* arow = A + (long)(m0 + l16) * HC1 + 2 * half;
    v8f accl = {}; v8f accr = {};
    for (int k0 = 0; k0 < HC1; k0 += 4) {
        v2f a;  a.x = arow[k0];  a.y = arow[k0 + 1];
        int kb = k0 + 2 * half;
        // B fragment (4x16 f32): row k striped over 16 lanes (col = n0+l16)
        v2f bL; bL.x = Wl[kb * C2 + n0 + l16]; bL.y = Wl[(kb + 1) * C2 + n0 + l16];
        v2f bR; bR.x = Wr[kb * C2 + n0 + l16]; bR.y = Wr[(kb + 1) * C2 + n0 + l16];
        accl = __builtin_amdgcn_wmma_f32_16x16x4_f32(false, a, false, bL, (short)0, accl, false, false);
        accr = __builtin_amdgcn_wmma_f32_16x16x4_f32(false, a, false, bR, (short)0, accr, false, false);
    }
#pragma unroll
    for (int r = 0; r < 8; ++r) {
        int row = m0 + r + 8 * half;   // C/D layout: VGPR r = rows r (lanes 0-15) and r+8 (lanes 16-31)
        int col = n0 + l16;
        xl2[(long)row * C2 + col] = accl[r] + bl[col];
        xr2[(long)row * C2 + col] = accr[r] + br[col];
    }
}

// -------- layer 2 edge logits: one wave per edge, single head, C2=64 --------
__global__ void edge_logits2_kernel(const float* __restrict__ xl, const float* __restrict__ xr,
                                    const int* __restrict__ src, const int* __restrict__ dst,
                                    const float* __restrict__ att, float* __restrict__ e2,
                                    int* __restrict__ emax_ord) {
    int warp = (blockIdx.x * blockDim.x + threadIdx.x) >> 5;
    int lane = threadIdx.x & 31;
    if (warp >= N_EDGES) return;
    int s = src[warp], d = dst[warp];
    float acc = 0.f;
    for (int c = lane; c < C2; c += 32)
        acc += lrelu(xl[(long)s * C2 + c] + xr[(long)d * C2 + c]) * att[c];
    for (int off = 16; off > 0; off >>= 1) acc += __shfl_xor(acc, off, 32);
    if (lane == 0) {
        e2[warp] = acc;
        atomicMax(emax_ord + d, f2ord(acc));
    }
}

__global__ void edge_scatter2_kernel(const float* __restrict__ xl,
                                     const int* __restrict__ src, const int* __restrict__ dst,
                                     const float* __restrict__ ee, const float* __restrict__ denom,
                                     float* __restrict__ out) {
    int warp = (blockIdx.x * blockDim.x + threadIdx.x) >> 5;
    int lane = threadIdx.x & 31;
    if (warp >= N_EDGES) return;
    int s = src[warp], d = dst[warp];
    float alpha = ee[warp] / (denom[d] + 1e-16f);
    for (int c = lane; c < C2; c += 32)
        atomicAdd(out + (long)d * C2 + c, alpha * xl[(long)s * C2 + c]);
}

__global__ void final_bias_kernel(const float* __restrict__ out2, const float* __restrict__ b2,
                                  float* __restrict__ y) {
    int i = blockIdx.x * blockDim.x + threadIdx.x;
    if (i >= N_NODES * C2) return;
    y[i] = out2[i] + b2[i % C2];
}

extern "C" void kernel_launch(void* const* d_in, const int* in_sizes, int n_in,
                              void* d_out, int out_size, void* d_ws, size_t ws_size,
                              hipStream_t stream) {
    (void)in_sizes; (void)n_in; (void)out_size; (void)ws_size;
    const float* x    = (const float*)d_in[0];
    const int*   esrc = (const int*)  d_in[1];
    const int*   edst = (const int*)  d_in[2];
    const float* Wl1  = (const float*)d_in[3];
    const float* bl1  = (const float*)d_in[4];
    const float* Wr1  = (const float*)d_in[5];
    const float* br1  = (const float*)d_in[6];
    const float* att1 = (const float*)d_in[7];
    const float* b1   = (const float*)d_in[8];
    const float* Wl2  = (const float*)d_in[9];
    const float* bl2  = (const float*)d_in[10];
    const float* Wr2  = (const float*)d_in[11];
    const float* br2  = (const float*)d_in[12];
    const float* att2 = (const float*)d_in[13];
    const float* b2   = (const float*)d_in[14];
    float* y = (float*)d_out;

    // workspace carve-up (256B aligned)
    char* ws = (char*)d_ws;
    size_t off = 0;
    auto carve = [&](size_t bytes) { char* p = ws + off; off = (off + bytes + 255) & ~(size_t)255; return p; };
    float* xl1    = (float*)carve((size_t)N_NODES * HC1 * 4);
    float* xr1    = (float*)carve((size_t)N_NODES * HC1 * 4);
    float* out1   = (float*)carve((size_t)N_NODES * HC1 * 4);   // becomes h after elu
    float* e1     = (float*)carve((size_t)N_EDGES * HEADS * 4);
    int*   emax1o = (int*)  carve((size_t)N_NODES * HEADS * 4);
    float* emax1f = (float*)carve((size_t)N_NODES * HEADS * 4);
    float* den1   = (float*)carve((size_t)N_NODES * HEADS * 4);
    float* xl2    = (float*)carve((size_t)N_NODES * C2 * 4);
    float* xr2    = (float*)carve((size_t)N_NODES * C2 * 4);
    float* e2     = (float*)carve((size_t)N_EDGES * 4);
    int*   emax2o = (int*)  carve((size_t)N_NODES * 4);
    float* emax2f = (float*)carve((size_t)N_NODES * 4);
    float* den2   = (float*)carve((size_t)N_NODES * 4);
    float* out2   = (float*)carve((size_t)N_NODES * C2 * 4);

    const int B = 256;
    auto grid = [](int n, int b) { return (n + b - 1) / b; };

    // ---- init accumulators (re-done every call: deterministic under graph replay) ----
    fill_f32_kernel<<<grid(N_NODES * HC1, B), B, 0, stream>>>(out1, 0.f, N_NODES * HC1);
    fill_f32_kernel<<<grid(N_NODES * C2, B), B, 0, stream>>>(out2, 0.f, N_NODES * C2);
    fill_i32_kernel<<<grid(N_NODES * HEADS, B), B, 0, stream>>>(emax1o, (int)0x80000000, N_NODES * HEADS);
    fill_f32_kernel<<<grid(N_NODES * HEADS, B), B, 0, stream>>>(den1, 0.f, N_NODES * HEADS);
    fill_i32_kernel<<<grid(N_NODES, B), B, 0, stream>>>(emax2o, (int)0x80000000, N_NODES);
    fill_f32_kernel<<<grid(N_NODES, B), B, 0, stream>>>(den2, 0.f, N_NODES);

    // ---- layer 1 ----
    gemm1_kernel<<<grid(N_NODES * HC1, B), B, 0, stream>>>(x, Wl1, bl1, Wr1, br1, xl1, xr1);
    edge_logits1_kernel<<<N_EDGES / 8, B, 0, stream>>>(xl1, xr1, esrc, edst, att1, e1, emax1o);
    finalize_max_kernel<<<grid(N_NODES * HEADS, B), B, 0, stream>>>(emax1o, emax1f, N_NODES * HEADS);
    edge_exp_kernel<<<grid(N_EDGES * HEADS, B), B, 0, stream>>>(edst, emax1f, e1, den1, N_EDGES, HEADS);
    edge_scatter1_kernel<<<N_EDGES / 8, B, 0, stream>>>(xl1, esrc, edst, e1, den1, out1);
    elu_bias1_kernel<<<grid(N_NODES * HC1, B), B, 0, stream>>>(out1, b1);

    // ---- layer 2 ----
    // 1250 m-tiles * 4 n-tiles = 5000 waves, 8 waves/block -> 625 blocks (exact; EXEC all-1s for WMMA)
    gemm2_wmma_kernel<<<625, B, 0, stream>>>(out1, Wl2, bl2, Wr2, br2, xl2, xr2);
    edge_logits2_kernel<<<N_EDGES / 8, B, 0, stream>>>(xl2, xr2, esrc, edst, att2, e2, emax2o);
    finalize_max_kernel<<<grid(N_NODES, B), B, 0, stream>>>(emax2o, emax2f, N_NODES);
    edge_exp_kernel<<<grid(N_EDGES, B), B, 0, stream>>>(edst, emax2f, e2, den2, N_EDGES, 1);
    edge_scatter2_kernel<<<N_EDGES / 8, B, 0, stream>>>(xl2, esrc, edst, e2, den2, out2);
    final_bias_kernel<<<grid(N_NODES * C2, B), B, 0, stream>>>(out2, b2, y);
}